// TokenLearner__22797686407652
// MI455X (gfx1250) — compile-verified
//
#include <hip/hip_runtime.h>
#include <hip/hip_bf16.h>

typedef float v2f __attribute__((ext_vector_type(2)));
typedef float v4f __attribute__((ext_vector_type(4)));
typedef float v8f __attribute__((ext_vector_type(8)));

#define BATCH 32
#define GRP   8
#define CIN   768
#define HWSZ  256
#define DIN   1536
#define KC    64        // K chunk staged in LDS
#define MG    64        // M rows per group (4 tiles of 16)
#define A_ST  68        // 68 mod 64 == 4 -> conflict-free b64 frag reads
#define BT_ST 68        // B stored transposed [n][k], same stride trick

__device__ __forceinline__ float gelu_erf(float v) {
    return 0.5f * v * (1.0f + erff(v * 0.70710678118654752f));
}

// One block per (b,g). 256 threads = 8 waves. Each wave: 4 M-tiles x 2 N-tiles.
// attn[b,g,hw] = W2[g] . gelu(W1[g] @ x[b] + b1[g]) + b2[g]
__global__ __launch_bounds__(256)
void tl_attn_kernel(const float* __restrict__ x,
                    const float* __restrict__ W1,
                    const float* __restrict__ b1,
                    const float* __restrict__ W2,
                    const float* __restrict__ b2,
                    float* __restrict__ attn)
{
    __shared__ __align__(16) float Alds[MG * A_ST];     // 17408 B, [m][k]
    __shared__ __align__(16) float Blds[HWSZ * BT_ST];  // 69632 B, [n][k] (transposed)

    const int b    = blockIdx.x >> 3;
    const int g    = blockIdx.x & 7;
    const int tid  = threadIdx.x;
    const int wave = tid >> 5;
    const int lane = tid & 31;
    const int ln16 = lane & 15;
    const int lh   = lane >> 4;

    const float* xb  = x  + (size_t)b * CIN * HWSZ;
    const float* w1g = W1 + (size_t)g * DIN * CIN;
    const float* b1g = b1 + (size_t)g * DIN;
    const float* w2g = W2 + (size_t)g * DIN;

    // columns (hw) owned by this lane
    const int n0 = wave * 16 + ln16;      // 0..127
    const int n1 = n0 + 128;              // 128..255

    float attnAcc0 = 0.0f, attnAcc1 = 0.0f;

    // cooperative-load geometry
    const int ldRowA = tid >> 2;          // 0..63   (A: 4 threads per row)
    const int ldColA = (tid & 3) * 16;    // 4 x float4 per thread
    const int ldRowB = (tid & 31) * 2;    // 0,2,..,62: each thread owns a k-pair
    const int ldColB = (tid >> 5) * 32;   // 8 segments of 32 columns (uniform per wave)

    for (int mg = 0; mg < DIN / MG; ++mg) {          // 24 M-groups
        v8f acc[4][2];
        #pragma unroll
        for (int i = 0; i < 4; ++i)
            #pragma unroll
            for (int j = 0; j < 2; ++j)
                acc[i][j] = (v8f){0.f,0.f,0.f,0.f,0.f,0.f,0.f,0.f};

        for (int kc = 0; kc < CIN / KC; ++kc) {      // 12 K-chunks
            __syncthreads();
            // stage A: W1 rows [mg*64,+64), cols [kc*64,+64)  (row-major)
            {
                const float* src = w1g + (size_t)(mg * MG + ldRowA) * CIN + kc * KC + ldColA;
                float* dst = &Alds[ldRowA * A_ST + ldColA];
                #pragma unroll
                for (int j = 0; j < 4; ++j)
                    *(v4f*)(dst + j * 4) = *(const v4f*)(src + j * 4);
            }
            // stage B transposed: Blds[n][k] = x[kc*64+k][n]
            // thread loads k-row pair -> packed b64 stores, banks {2l,2l+1}: conflict-free
            {
                const float* src0 = xb + (size_t)(kc * KC + ldRowB) * HWSZ + ldColB;
                const float* src1 = src0 + HWSZ;   // next k row
                #pragma unroll
                for (int j = 0; j < 8; ++j) {
                    v4f v0 = *(const v4f*)(src0 + j * 4);
                    v4f v1 = *(const v4f*)(src1 + j * 4);
                    const int n = ldColB + j * 4;
                    #pragma unroll
                    for (int e = 0; e < 4; ++e) {
                        v2f p = {v0[e], v1[e]};
                        *(v2f*)&Blds[(n + e) * BT_ST + ldRowB] = p;  // ds_store_b64
                    }
                }
            }
            __syncthreads();

            #pragma unroll 4
            for (int k4 = 0; k4 < KC / 4; ++k4) {    // 16 x (K=4) steps
                const int ak = k4 * 4 + 2 * lh;      // frag K = 2*lh + {0,1}
                v2f a0 = *(const v2f*)&Alds[(0 * 16 + ln16) * A_ST + ak];
                v2f a1 = *(const v2f*)&Alds[(1 * 16 + ln16) * A_ST + ak];
                v2f a2 = *(const v2f*)&Alds[(2 * 16 + ln16) * A_ST + ak];
                v2f a3 = *(const v2f*)&Alds[(3 * 16 + ln16) * A_ST + ak];
                v2f bf0 = *(const v2f*)&Blds[n0 * BT_ST + ak];   // single b64
                v2f bf1 = *(const v2f*)&Blds[n1 * BT_ST + ak];
                acc[0][0] = __builtin_amdgcn_wmma_f32_16x16x4_f32(false, a0, false, bf0, (short)0, acc[0][0], false, false);
                acc[1][0] = __builtin_amdgcn_wmma_f32_16x16x4_f32(false, a1, false, bf0, (short)0, acc[1][0], false, false);
                acc[2][0] = __builtin_amdgcn_wmma_f32_16x16x4_f32(false, a2, false, bf0, (short)0, acc[2][0], false, false);
                acc[3][0] = __builtin_amdgcn_wmma_f32_16x16x4_f32(false, a3, false, bf0, (short)0, acc[3][0], false, false);
                acc[0][1] = __builtin_amdgcn_wmma_f32_16x16x4_f32(false, a0, false, bf1, (short)0, acc[0][1], false, false);
                acc[1][1] = __builtin_amdgcn_wmma_f32_16x16x4_f32(false, a1, false, bf1, (short)0, acc[1][1], false, false);
                acc[2][1] = __builtin_amdgcn_wmma_f32_16x16x4_f32(false, a2, false, bf1, (short)0, acc[2][1], false, false);
                acc[3][1] = __builtin_amdgcn_wmma_f32_16x16x4_f32(false, a3, false, bf1, (short)0, acc[3][1], false, false);
            }
        }

        // fused epilogue: bias + exact GELU + W2 reduction over 64 rows
        #pragma unroll
        for (int mt = 0; mt < 4; ++mt) {
            const int mbase = mg * MG + mt * 16 + lh * 8;   // C VGPR r -> row r / r+8
            v4f bi0 = *(const v4f*)&b1g[mbase];
            v4f bi1 = *(const v4f*)&b1g[mbase + 4];
            v4f w20 = *(const v4f*)&w2g[mbase];
            v4f w21 = *(const v4f*)&w2g[mbase + 4];
            #pragma unroll
            for (int r = 0; r < 4; ++r) {
                attnAcc0 += w20[r] * gelu_erf(acc[mt][0][r] + bi0[r]);
                attnAcc1 += w20[r] * gelu_erf(acc[mt][1][r] + bi0[r]);
            }
            #pragma unroll
            for (int r = 0; r < 4; ++r) {
                attnAcc0 += w21[r] * gelu_erf(acc[mt][0][r + 4] + bi1[r]);
                attnAcc1 += w21[r] * gelu_erf(acc[mt][1][r + 4] + bi1[r]);
            }
        }
    }

    // column n gets contributions from both lane halves (wave32 shuffle)
    attnAcc0 += __shfl_xor(attnAcc0, 16, 32);
    attnAcc1 += __shfl_xor(attnAcc1, 16, 32);
    if (lh == 0) {
        const float bb = b2[g];
        float* arow = attn + (size_t)(b * GRP + g) * HWSZ;
        arow[n0] = attnAcc0 + bb;
        arow[n1] = attnAcc1 + bb;
    }
}

// out[b,c,g] = (1/256) * sum_hw x[b,c,hw] * attn[b,g,hw]
__global__ __launch_bounds__(256)
void tl_pool_kernel(const float* __restrict__ x,
                    const float* __restrict__ attn,
                    float* __restrict__ out)
{
    __shared__ __align__(16) float att[GRP * HWSZ];   // 8 KB
    const int b   = blockIdx.x;
    const int tid = threadIdx.x;

    for (int i = tid; i < GRP * HWSZ; i += 256)
        att[i] = attn[(size_t)b * GRP * HWSZ + i];
    __syncthreads();

    const int g  = tid & 7;
    const int c0 = tid >> 3;                          // 0..31
    const float* ag = &att[g * HWSZ];

    for (int ci = 0; ci < CIN / 32; ++ci) {           // 24 channels/thread
        const int c = ci * 32 + c0;
        const float* xr = x + ((size_t)b * CIN + c) * HWSZ;
        float acc = 0.0f;
        #pragma unroll 8
        for (int h4 = 0; h4 < HWSZ / 4; ++h4) {
            v4f xv = *(const v4f*)(xr + h4 * 4);
            acc += xv.x * ag[h4 * 4 + 0] + xv.y * ag[h4 * 4 + 1]
                 + xv.z * ag[h4 * 4 + 2] + xv.w * ag[h4 * 4 + 3];
        }
        out[((size_t)b * CIN + c) * GRP + g] = acc * (1.0f / 256.0f);
    }
}

extern "C" void kernel_launch(void* const* d_in, const int* in_sizes, int n_in,
                              void* d_out, int out_size, void* d_ws, size_t ws_size,
                              hipStream_t stream) {
    const float* x  = (const float*)d_in[0];
    const float* W1 = (const float*)d_in[1];
    const float* b1 = (const float*)d_in[2];
    const float* W2 = (const float*)d_in[3];
    const float* b2 = (const float*)d_in[4];
    float* out  = (float*)d_out;
    float* attn = (float*)d_ws;   // B*G*HW floats = 256 KB scratch

    tl_attn_kernel<<<BATCH * GRP, 256, 0, stream>>>(x, W1, b1, W2, b2, attn);
    tl_pool_kernel<<<BATCH, 256, 0, stream>>>(x, attn, out);
}